// PrRoIPool2D_30666066493717
// MI455X (gfx1250) — compile-verified
//
#include <hip/hip_runtime.h>

#define POOLED 7
#define CHN 256
#define HF 48
#define WF 48
#define RR 300
#define SCALE 0.0625f

typedef __attribute__((ext_vector_type(2))) float v2f;
typedef __attribute__((ext_vector_type(8))) float v8f;

// Integral of unit hat basis centered at g, from -inf to x.
__device__ __forceinline__ float hat_int(float x, float g) {
    float t = x - (g - 1.0f);
    float r;
    if (t <= 0.0f)      r = 0.0f;
    else if (t <= 1.0f) r = 0.5f * t * t;
    else if (t <= 2.0f) { float u = 2.0f - t; r = 1.0f - 0.5f * u * u; }
    else                r = 1.0f;
    return r;
}

// Intra-wave LDS ordering: LDS ops of one wave execute in order on CDNA5,
// so we only need to stop the compiler from reordering/caching across this.
__device__ __forceinline__ void wave_lds_fence() {
    __builtin_amdgcn_wave_barrier();
    asm volatile("" ::: "memory");
}

// One block = (channel-group, roi). 8 waves; each wave handles 4 channels.
// Per channel: O(16x16, pad of 7x7) = Wy(16x48,pad) @ F(48x48) @ Wx^T(48x16,pad)
// via V_WMMA_F32_16X16X4_F32 (exact fp32, K=4 per issue).
__global__ __launch_bounds__(256) void prroi_wmma_kernel(
    const float* __restrict__ feat,   // [8,256,48,48]
    const float* __restrict__ rois,   // [300,5]
    float* __restrict__ out)          // [300,256,7,7]
{
    __shared__ float sWy[POOLED * WF];    // 7 x 48 y-axis weights
    __shared__ float sWx[POOLED * WF];    // 7 x 48 x-axis weights
    __shared__ float sT[8][16 * 49];      // per-wave 16x48 T tile, stride 49

    const int r    = blockIdx.y;
    const int cgrp = blockIdx.x;          // 0..7 -> 32 channels
    const int tid  = threadIdx.x;
    const int lane = tid & 31;            // wave32
    const int wave = tid >> 5;
    const int hn   = lane >> 4;           // half-wave (K phase in A/B fragments)
    const int mn   = lane & 15;           // M (A rows) / N (cols) index

    // ---- ROI parameters (uniform per block) ----
    const float x1 = rois[r * 5 + 1] * SCALE;
    const float y1 = rois[r * 5 + 2] * SCALE;
    const float x2 = rois[r * 5 + 3] * SCALE;
    const float y2 = rois[r * 5 + 4] * SCALE;
    const int   bi = (int)rois[r * 5 + 0];
    const float roi_w = fmaxf(x2 - x1, 0.0f);
    const float roi_h = fmaxf(y2 - y1, 0.0f);
    const float bin_w = roi_w * (1.0f / POOLED);
    const float bin_h = roi_h * (1.0f / POOLED);
    const float area  = bin_w * bin_h;
    const float inv_area = (area > 0.0f) ? (1.0f / area) : 0.0f;

    // ---- Per-axis hat-integral weights into LDS (672 values, 3 passes) ----
    for (int idx = tid; idx < 2 * POOLED * WF; idx += 256) {
        int axis = idx / (POOLED * WF);
        int rem  = idx - axis * (POOLED * WF);
        int p    = rem / WF;
        int j    = rem - p * WF;
        float a0, bw;
        if (axis == 0) { a0 = y1 + p * bin_h; bw = bin_h; }
        else           { a0 = x1 + p * bin_w; bw = bin_w; }
        float w = hat_int(a0 + bw, (float)j) - hat_int(a0, (float)j);
        if (axis == 0) sWy[rem] = w;
        else           sWx[rem] = w;
    }
    __syncthreads();   // real cross-wave barrier: weights are shared

    // ---- Resident fragments (reused for all channels of this wave) ----
    // A layout (16x4 f32): lane m=mn; v0 = A[m][k0+2h], v1 = A[m][k0+2h+1]
    v2f a1[12];   // stage-1 A = Wy, rows >= 7 zero-padded
    v2f b2[12];   // stage-2 B = Wx^T: B[i][q] = Wx[q][i], cols q >= 7 zero-padded
#pragma unroll
    for (int kk = 0; kk < 12; ++kk) {
        int k0 = 4 * kk + 2 * hn;
        float wy0 = (mn < POOLED) ? sWy[mn * WF + k0]     : 0.0f;
        float wy1 = (mn < POOLED) ? sWy[mn * WF + k0 + 1] : 0.0f;
        float wx0 = (mn < POOLED) ? sWx[mn * WF + k0]     : 0.0f;
        float wx1 = (mn < POOLED) ? sWx[mn * WF + k0 + 1] : 0.0f;
        a1[kk].x = wy0; a1[kk].y = wy1;
        b2[kk].x = wx0; b2[kk].y = wx1;
    }

    float* Tw = sT[wave];

    for (int cc = 0; cc < 4; ++cc) {
        const int c = cgrp * 32 + wave * 4 + cc;
        const float* __restrict__ F =
            feat + ((size_t)bi * CHN + (size_t)c) * (HF * WF);

        // Prefetch next 4 channel slices (9216 floats = 288 cachelines) into
        // WGP-level caches while this channel's WMMAs run.
        if (cc < 3) {
            const float* Fn = F + HF * WF;
#pragma unroll
            for (int pb = 0; pb < 9; ++pb) {
                __builtin_prefetch(Fn + (pb * 32 + lane) * 32, 0, 3);
            }
        }

        // ---- Stage 1: T(16x48) = Wy_pad @ F ; 3 N-tiles x 12 K-steps ----
#pragma unroll
        for (int it = 0; it < 3; ++it) {
            v8f acc = {};
            const int col = it * 16 + mn;
#pragma unroll
            for (int kk = 0; kk < 12; ++kk) {
                const int row = 4 * kk + 2 * hn;
                v2f b;
                b.x = F[row * WF + col];
                b.y = F[(row + 1) * WF + col];
                acc = __builtin_amdgcn_wmma_f32_16x16x4_f32(
                    false, a1[kk], false, b, (short)0, acc, false, false);
            }
            // C/D layout: lane holds rows v+8*hn at column col
#pragma unroll
            for (int v = 0; v < 8; ++v) {
                Tw[(v + 8 * hn) * 49 + col] = acc[v];
            }
        }
        wave_lds_fence();   // publish T across lanes (intra-wave, in-order LDS)

        // ---- Stage 2: O(16x16) = T @ Wx^T ; 12 K-steps ----
        v8f acc2 = {};
#pragma unroll
        for (int kk = 0; kk < 12; ++kk) {
            const int k0 = 4 * kk + 2 * hn;
            v2f a;
            a.x = Tw[mn * 49 + k0];       // consecutive -> ds_load_b64
            a.y = Tw[mn * 49 + k0 + 1];
            acc2 = __builtin_amdgcn_wmma_f32_16x16x4_f32(
                false, a, false, b2[kk], (short)0, acc2, false, false);
        }
        wave_lds_fence();   // T reads done before next channel overwrites

        // ---- Store 7x7 bin, scaled by 1/area (lanes 0..6, rows 0..6) ----
        if (hn == 0 && mn < POOLED) {
            float* o = out + ((size_t)r * CHN + (size_t)c) * (POOLED * POOLED) + mn;
#pragma unroll
            for (int v = 0; v < POOLED; ++v) {
                o[v * POOLED] = acc2[v] * inv_area;
            }
        }
    }
}

extern "C" void kernel_launch(void* const* d_in, const int* in_sizes, int n_in,
                              void* d_out, int out_size, void* d_ws, size_t ws_size,
                              hipStream_t stream) {
    const float* feat = (const float*)d_in[0];
    const float* rois = (const float*)d_in[1];
    float* out = (float*)d_out;
    dim3 grid(8, RR);   // 8 channel groups x 300 rois = 2400 blocks
    dim3 block(256);    // 8 wave32
    hipLaunchKernelGGL(prroi_wmma_kernel, grid, block, 0, stream, feat, rois, out);
}